// SinkhornDistance_1065151889712
// MI455X (gfx1250) — compile-verified
//
#include <hip/hip_runtime.h>

#define PDIM 1024
#define NSL 16
#define EPSV 0.001f
#define LN2F 0.6931471805599453f
// (1/EPS) * log2(e): work in log2 domain so exp -> single v_exp_f32
#define SCALE 1442.6950408889634f

typedef float v8f __attribute__((ext_vector_type(8)));
typedef float v2f __attribute__((ext_vector_type(2)));

// Raw v_exp_f32: argument is max-subtracted (<= 0) in all LSE uses, so
// flush-to-zero below 2^-126 is numerically irrelevant (sum >= 1).
__device__ __forceinline__ float fast_exp2(float x) {
    return __builtin_amdgcn_exp2f(x);
}

// ---------------------------------------------------------------------------
// u initialization: u0 = ones (v0 cancels algebraically, never needed)
// ---------------------------------------------------------------------------
__global__ __launch_bounds__(256) void sinkhorn_init_u(float* __restrict__ u) {
    u[blockIdx.x * 256 + threadIdx.x] = 1.0f;
}

// ---------------------------------------------------------------------------
// Phase A (v update): column LSE over rows i.
//   v_j = EPS * ( log(nu_j + 1e-8) - LSE_i( (u_i - C_ij)/EPS ) )
// Two-pass stabilized LSE. Pass 2 accumulates exp values with
// V_WMMA_F32_16X16X4_F32 against an all-ones A matrix: D += ones * E,
// giving exact fp32 column sums on the matrix pipe (co-executes with
// v_exp_f32 on the TRANS pipe). grid = 16 slices * 4 column blocks.
// ---------------------------------------------------------------------------
__global__ __launch_bounds__(256) void sinkhorn_v_update(
    const float* __restrict__ C, const float* __restrict__ nu,
    const float* __restrict__ u, float* __restrict__ v) {
    __shared__ float us[PDIM];   // u_i * SCALE (log2-scaled)
    __shared__ float sm[256];    // per-column max
    const int tid  = threadIdx.x;
    const int n    = blockIdx.x >> 2;
    const int cblk = (blockIdx.x & 3) << 8;
    const float* Cn = C + (size_t)n * PDIM * PDIM;

    for (int i = tid; i < PDIM; i += 256) us[i] = u[n * PDIM + i] * SCALE;
    __syncthreads();

    // pass 1: per-column max of a_ij = (u_i - C_ij)*SCALE  (coalesced rows)
    const int col = cblk + tid;
    float m = -3.4e38f;
    #pragma unroll 4
    for (int i = 0; i < PDIM; ++i) {
        float a = fmaf(-Cn[(size_t)i * PDIM + col], SCALE, us[i]);
        m = fmaxf(m, a);
    }
    sm[tid] = m;
    __syncthreads();

    // pass 2: WMMA-accumulated column sums of exp2(a - m).
    // Each wave owns two 16-column tiles; 4 rows consumed per WMMA.
    const int wave = tid >> 5;
    const int lane = tid & 31;
    const int l16  = lane & 15;
    const int kof  = lane >> 4;             // half-wave row slot
    const int colA = cblk + wave * 32 + l16;
    const int colB = colA + 16;
    const float mA = sm[wave * 32 + l16];
    const float mB = sm[wave * 32 + 16 + l16];

    v2f ones; ones[0] = 1.0f; ones[1] = 1.0f;   // A = all-ones (layout-invariant)
    v8f accA = {}; v8f accB = {};
    #pragma unroll 2
    for (int r0 = 0; r0 < PDIM; r0 += 4) {
        const int r = r0 + kof;              // each (vgpr, half) slot -> unique row
        const float u0 = us[r];
        const float u1 = us[r + 2];
        const size_t o0 = (size_t)r * PDIM;
        const size_t o1 = (size_t)(r + 2) * PDIM;
        v2f bA, bB;
        bA[0] = fast_exp2(fmaf(-Cn[o0 + colA], SCALE, u0) - mA);
        bA[1] = fast_exp2(fmaf(-Cn[o1 + colA], SCALE, u1) - mA);
        bB[0] = fast_exp2(fmaf(-Cn[o0 + colB], SCALE, u0) - mB);
        bB[1] = fast_exp2(fmaf(-Cn[o1 + colB], SCALE, u1) - mB);
        accA = __builtin_amdgcn_wmma_f32_16x16x4_f32(
            false, ones, false, bA, (short)0, accA, false, false);
        accB = __builtin_amdgcn_wmma_f32_16x16x4_f32(
            false, ones, false, bB, (short)0, accB, false, false);
    }
    // With A = ones, every D row equals the column sums; element 0 of any lane
    // holds the sum for column (lane & 15) of the tile.
    if (lane < 16) {
        const float lseA = (mA + log2f(accA[0])) * LN2F;
        const float lseB = (mB + log2f(accB[0])) * LN2F;
        v[n * PDIM + colA] = EPSV * (__logf(nu[n * PDIM + colA] + 1e-8f) - lseA);
        v[n * PDIM + colB] = EPSV * (__logf(nu[n * PDIM + colB] + 1e-8f) - lseB);
    }
}

// ---------------------------------------------------------------------------
// Phase B (u update): row LSE over columns j, one wave32 per row.
//   u_i = EPS * ( log(mu_i + 1e-8) - LSE_j( (v_j - C_ij)/EPS ) )
// Row cached in 32 VGPRs via coalesced b128 loads; exact two-pass LSE with
// shuffle-xor max/sum combines. grid = 16*1024 rows / 8 waves per block.
// ---------------------------------------------------------------------------
__global__ __launch_bounds__(256) void sinkhorn_u_update(
    const float* __restrict__ C, const float* __restrict__ mu,
    const float* __restrict__ v, float* __restrict__ u) {
    const int gw   = (blockIdx.x << 3) + (threadIdx.x >> 5);  // global row
    const int lane = threadIdx.x & 31;
    const int n    = gw >> 10;
    const int row  = gw & 1023;
    const float* Cr = C + ((size_t)(n * PDIM + row)) * PDIM;
    const float* vn = v + n * PDIM;

    float a[32];
    float m = -3.4e38f;
    #pragma unroll
    for (int t = 0; t < 8; ++t) {
        const int j = t * 128 + lane * 4;           // wave covers 128 consecutive
        const float4 c4 = *(const float4*)(Cr + j);
        const float4 v4 = *(const float4*)(vn + j);
        a[4*t+0] = (v4.x - c4.x) * SCALE;
        a[4*t+1] = (v4.y - c4.y) * SCALE;
        a[4*t+2] = (v4.z - c4.z) * SCALE;
        a[4*t+3] = (v4.w - c4.w) * SCALE;
        m = fmaxf(m, fmaxf(fmaxf(a[4*t+0], a[4*t+1]), fmaxf(a[4*t+2], a[4*t+3])));
    }
    #pragma unroll
    for (int off = 16; off > 0; off >>= 1)
        m = fmaxf(m, __shfl_xor(m, off, 32));

    float s = 0.0f;
    #pragma unroll
    for (int k = 0; k < 32; ++k) s += fast_exp2(a[k] - m);
    #pragma unroll
    for (int off = 16; off > 0; off >>= 1)
        s += __shfl_xor(s, off, 32);

    if (lane == 0) {
        const float lse = (m + log2f(s)) * LN2F;
        u[n * PDIM + row] = EPSV * (__logf(mu[n * PDIM + row] + 1e-8f) - lse);
    }
}

// ---------------------------------------------------------------------------
// Finalize: pi = exp((u_i + v_j - C_ij)/EPS), per-block partial of sum(pi*C).
// grid = 4096 blocks (256 per slice), 4 elements/thread, b128 stores.
// ---------------------------------------------------------------------------
__global__ __launch_bounds__(256) void sinkhorn_finalize(
    const float* __restrict__ C, const float* __restrict__ u,
    const float* __restrict__ v, float* __restrict__ pi,
    float* __restrict__ partial) {
    __shared__ float red[256];
    const int tid = threadIdx.x;
    const int n   = blockIdx.x >> 8;
    const int blk = blockIdx.x & 255;
    const int e   = blk * 4096 + tid * 4;        // element offset within slice
    const size_t g = (size_t)n * PDIM * PDIM + e;
    const int i = e >> 10;
    const int j = e & 1023;

    const float ui = u[n * PDIM + i];
    const float4 v4 = *(const float4*)(v + n * PDIM + j);
    const float4 c4 = *(const float4*)(C + g);
    float4 p;
    p.x = fast_exp2((ui + v4.x - c4.x) * SCALE);
    p.y = fast_exp2((ui + v4.y - c4.y) * SCALE);
    p.z = fast_exp2((ui + v4.z - c4.z) * SCALE);
    p.w = fast_exp2((ui + v4.w - c4.w) * SCALE);
    *(float4*)(pi + g) = p;

    red[tid] = p.x * c4.x + p.y * c4.y + p.z * c4.z + p.w * c4.w;
    __syncthreads();
    #pragma unroll
    for (int off = 128; off > 0; off >>= 1) {
        if (tid < off) red[tid] += red[tid + off];
        __syncthreads();
    }
    if (tid == 0) partial[blockIdx.x] = red[0];
}

// Deterministic fixed-order cost reduction: block b reduces slice b's 256 partials.
__global__ __launch_bounds__(256) void sinkhorn_cost(
    const float* __restrict__ partial, float* __restrict__ cost) {
    __shared__ float red[256];
    const int tid = threadIdx.x;
    red[tid] = partial[blockIdx.x * 256 + tid];
    __syncthreads();
    #pragma unroll
    for (int off = 128; off > 0; off >>= 1) {
        if (tid < off) red[tid] += red[tid + off];
        __syncthreads();
    }
    if (tid == 0) cost[blockIdx.x] = red[0];
}

// ---------------------------------------------------------------------------
extern "C" void kernel_launch(void* const* d_in, const int* in_sizes, int n_in,
                              void* d_out, int out_size, void* d_ws, size_t ws_size,
                              hipStream_t stream) {
    (void)in_sizes; (void)n_in; (void)out_size; (void)ws_size;
    const float* mu = (const float*)d_in[0];
    const float* nu = (const float*)d_in[1];
    const float* C  = (const float*)d_in[2];
    float* cost = (float*)d_out;        // 16 costs
    float* pi   = cost + 16;            // 16 * 1024 * 1024 transport plan
    float* u    = (float*)d_ws;         // 16*1024
    float* v    = u + NSL * PDIM;       // 16*1024
    float* part = v + NSL * PDIM;       // 4096 block partials

    sinkhorn_init_u<<<64, 256, 0, stream>>>(u);
    for (int it = 0; it < 100; ++it) {
        sinkhorn_v_update<<<64,   256, 0, stream>>>(C, nu, u, v);
        sinkhorn_u_update<<<2048, 256, 0, stream>>>(C, mu, v, u);
    }
    sinkhorn_finalize<<<4096, 256, 0, stream>>>(C, u, v, pi, part);
    sinkhorn_cost<<<16, 256, 0, stream>>>(part, cost);
}